// sp_attn_head_13297218748804
// MI455X (gfx1250) — compile-verified
//
#include <hip/hip_runtime.h>
#include <math.h>

typedef __attribute__((ext_vector_type(2))) float v2f;
typedef __attribute__((ext_vector_type(8))) float v8f;

#define F_IN 128
#define HID 64
#define LRELU 0.2f
#define CHUNK 256

// ---------- monotone ordered-uint encoding for float atomic max ----------
__device__ __forceinline__ unsigned f32_ord(float f) {
  unsigned u = __float_as_uint(f);
  return (u & 0x80000000u) ? ~u : (u | 0x80000000u);
}
__device__ __forceinline__ float ord_f32(unsigned k) {
  unsigned u = (k & 0x80000000u) ? (k & 0x7FFFFFFFu) : ~k;
  return __uint_as_float(u);
}
#define ORD_NEG_INF 0x007FFFFFu  // f32_ord(-inf)

// ---------- 1) seq_fts = seq @ W via V_WMMA_F32_16X16X4_F32 ----------
// Block = 128 threads = 4 waves. Each block: 16 rows; wave w does cols [16w,16w+16).
// A 16x4 f32 frag: lane<16 -> (M=lane, K=kk+0 / kk+1), lane>=16 -> (M=lane-16, K=kk+2 / kk+3)
// B 4x16 f32 frag: same K split, N = lane&15.
// C/D: VGPR v -> M = v + (lane<16 ? 0 : 8), N = lane&15.
__global__ void gemm_wmma_kernel(const float* __restrict__ A,   // [N, F_IN]
                                 const float* __restrict__ W,   // [F_IN, HID]
                                 float* __restrict__ C,         // [N, HID]
                                 int N) {
  const int lane = threadIdx.x & 31;
  const int wave = threadIdx.x >> 5;          // 0..3 column tile
  const int r0   = blockIdx.x * 16;
  const int col0 = wave * 16;
  const int m    = lane & 15;
  const int koff = (lane >> 4) << 1;          // 0 or 2

  int rowA = r0 + m;
  if (rowA >= N) rowA = N - 1;                // clamp: keep EXEC all-ones for WMMA
  const float* aptr = A + (size_t)rowA * F_IN + koff;
  const float* bptr = W + (size_t)koff * HID + col0 + m;

  v8f acc = {};
#pragma unroll
  for (int kk = 0; kk < F_IN; kk += 4) {
    v2f a = *(const v2f*)(aptr + kk);         // (kk+koff, kk+koff+1) : consecutive in K
    v2f b;
    b.x = bptr[(size_t)kk * HID];
    b.y = bptr[(size_t)(kk + 1) * HID];
    acc = __builtin_amdgcn_wmma_f32_16x16x4_f32(
        /*neg_a=*/false, a, /*neg_b=*/false, b,
        /*c_mod=*/(short)0, acc, /*reuse_a=*/false, /*reuse_b=*/false);
  }

#pragma unroll
  for (int v = 0; v < 8; ++v) {
    int row = r0 + v + ((lane >> 4) << 3);
    if (row < N) C[(size_t)row * HID + col0 + (lane & 15)] = acc[v];
  }
}

// ---------- 2) f1 = seq_fts @ a1 + b1 ; f2 = seq_fts @ a2 + b2 ----------
__global__ void fvec_kernel(const float* __restrict__ sf,
                            const float* __restrict__ a1, const float* __restrict__ b1,
                            const float* __restrict__ a2, const float* __restrict__ b2,
                            float* __restrict__ f1, float* __restrict__ f2, int N) {
  int r = blockIdx.x * blockDim.x + threadIdx.x;
  if (r >= N) return;
  const float* p = sf + (size_t)r * HID;
  float s1 = 0.f, s2 = 0.f;
#pragma unroll
  for (int h = 0; h < HID; h += 4) {
    float4 v = *(const float4*)(p + h);
    float4 x = *(const float4*)(a1 + h);
    float4 y = *(const float4*)(a2 + h);
    s1 += v.x * x.x + v.y * x.y + v.z * x.z + v.w * x.w;
    s2 += v.x * y.x + v.y * y.y + v.z * y.z + v.w * y.w;
  }
  f1[r] = s1 + b1[0];
  f2[r] = s2 + b2[0];
}

// ---------- 3) init: zero accumulation target, -inf row max, zero denom ----------
__global__ void init_kernel(float* __restrict__ out, unsigned* __restrict__ rowmax,
                            float* __restrict__ denom, int N) {
  long i = (long)blockIdx.x * blockDim.x + threadIdx.x;
  if (i < (long)N * HID) out[i] = 0.f;
  if (i < N) { rowmax[i] = ORD_NEG_INF; denom[i] = 0.f; }
}

// ---------- 4) per-edge leaky-relu logits + segmented max ----------
__global__ void edge_logits_max_kernel(const int* __restrict__ er, const int* __restrict__ ec,
                                       const float* __restrict__ ev,
                                       const float* __restrict__ f1, const float* __restrict__ f2,
                                       float* __restrict__ logits, unsigned* __restrict__ rowmax,
                                       long E) {
  long e = (long)blockIdx.x * blockDim.x + threadIdx.x;
  if (e >= E) return;
  int r = er[e], c = ec[e];
  float lg = ev[e] * (f1[r] + f2[c]);
  lg = (lg > 0.f) ? lg : LRELU * lg;
  logits[e] = lg;
  atomicMax(rowmax + r, f32_ord(lg));
}

// ---------- 5) z = exp(logit - rowmax); segmented sum ----------
__global__ void edge_exp_sum_kernel(const int* __restrict__ er, float* __restrict__ logits,
                                    const unsigned* __restrict__ rowmax,
                                    float* __restrict__ denom, long E) {
  long e = (long)blockIdx.x * blockDim.x + threadIdx.x;
  if (e >= E) return;
  int r = er[e];
  float z = expf(logits[e] - ord_f32(rowmax[r]));
  logits[e] = z;
  atomicAdd(denom + r, z);
}

// ---------- 6) out[row] += (z/denom[row]) * seq_fts[col] ----------
// One wave walks CHUNK consecutive edges; lane owns hidden channels 2*lane, 2*lane+1.
// edge_row is sorted -> accumulate in registers, flush via atomics on row change.
__global__ void edge_accum_kernel(const int* __restrict__ er, const int* __restrict__ ec,
                                  const float* __restrict__ z, const float* __restrict__ denom,
                                  const float* __restrict__ sf, float* __restrict__ out, long E) {
  long wid = ((long)blockIdx.x * blockDim.x + threadIdx.x) >> 5;
  int lane = threadIdx.x & 31;
  long e0 = wid * CHUNK;
  if (e0 >= E) return;
  long e1 = e0 + CHUNK;
  if (e1 > E) e1 = E;

  int cur = -1;
  float acc0 = 0.f, acc1 = 0.f;
  for (long e = e0; e < e1; ++e) {
    int r = er[e];
    if (r != cur) {
      if (cur >= 0) {
        atomicAdd(out + (size_t)cur * HID + lane * 2,     acc0);
        atomicAdd(out + (size_t)cur * HID + lane * 2 + 1, acc1);
      }
      cur = r; acc0 = 0.f; acc1 = 0.f;
    }
    int c = ec[e];
    float coef = z[e] / denom[r];
    v2f v = *(const v2f*)(sf + (size_t)c * HID + lane * 2);
    acc0 += coef * v.x;
    acc1 += coef * v.y;
  }
  if (cur >= 0) {
    atomicAdd(out + (size_t)cur * HID + lane * 2,     acc0);
    atomicAdd(out + (size_t)cur * HID + lane * 2 + 1, acc1);
  }
}

// ---------- 7) out = elu(out + bias_zero) in place ----------
__global__ void elu_kernel(float* __restrict__ out, const float* __restrict__ bias, long total) {
  long i = (long)blockIdx.x * blockDim.x + threadIdx.x;
  if (i >= total) return;
  float x = out[i] + bias[i & (HID - 1)];
  out[i] = (x > 0.f) ? x : (expf(x) - 1.f);
}

extern "C" void kernel_launch(void* const* d_in, const int* in_sizes, int n_in,
                              void* d_out, int out_size, void* d_ws, size_t ws_size,
                              hipStream_t stream) {
  const float* seq  = (const float*)d_in[0];
  const int*   er   = (const int*)d_in[1];
  const int*   ec   = (const int*)d_in[2];
  const float* ev   = (const float*)d_in[3];
  const float* W    = (const float*)d_in[4];
  const float* a1   = (const float*)d_in[5];
  const float* b1   = (const float*)d_in[6];
  const float* a2   = (const float*)d_in[7];
  const float* b2   = (const float*)d_in[8];
  const float* bias = (const float*)d_in[9];

  int  N = in_sizes[0] / F_IN;
  long E = in_sizes[1];
  float* out = (float*)d_out;

  // workspace layout (floats)
  float*    sf     = (float*)d_ws;                     // N*HID
  float*    f1     = sf + (size_t)N * HID;             // N
  float*    f2     = f1 + N;                           // N
  unsigned* rowmax = (unsigned*)(f2 + N);              // N
  float*    denom  = (float*)(rowmax + N);             // N
  float*    zbuf   = denom + N;                        // E

  long total = (long)N * HID;

  int tiles = (N + 15) / 16;
  gemm_wmma_kernel<<<tiles, 128, 0, stream>>>(seq, W, sf, N);
  fvec_kernel<<<(N + 255) / 256, 256, 0, stream>>>(sf, a1, b1, a2, b2, f1, f2, N);
  init_kernel<<<(int)((total + 255) / 256), 256, 0, stream>>>(out, rowmax, denom, N);
  edge_logits_max_kernel<<<(int)((E + 255) / 256), 256, 0, stream>>>(er, ec, ev, f1, f2,
                                                                     zbuf, rowmax, E);
  edge_exp_sum_kernel<<<(int)((E + 255) / 256), 256, 0, stream>>>(er, zbuf, rowmax, denom, E);
  long nwaves   = (E + CHUNK - 1) / CHUNK;
  long nthreads = nwaves * 32;
  edge_accum_kernel<<<(int)((nthreads + 255) / 256), 256, 0, stream>>>(er, ec, zbuf, denom,
                                                                       sf, out, E);
  elu_kernel<<<(int)((total + 255) / 256), 256, 0, stream>>>(out, bias, total);
}